// MeshTorchLayer_82059645157414
// MI455X (gfx1250) — compile-verified
//
#include <hip/hip_runtime.h>
#include <hip/hip_bf16.h>

#define UNITS 256
#define LAYERS 256
#define BATCH 4096
#define MZIS 128
#define PI_F 3.14159265358979323846f

typedef __attribute__((ext_vector_type(2))) float v2f;
typedef __attribute__((ext_vector_type(8))) float v8f;

// ---------------------------------------------------------------------------
// Kernel 0: per-(layer, unit) MZI coefficients D,O (complex), with masking
// (masked MZI pinned to theta=phi=pi -> identity transfer) and the odd-layer
// roll-by-1 baked in.  coef[l*256+n] = {Dre, Dim, Ore, Oim}.
// ---------------------------------------------------------------------------
__global__ __launch_bounds__(256) void mesh_coef_kernel(
    const float* __restrict__ theta, const float* __restrict__ phi,
    const int* __restrict__ mask, float4* __restrict__ coef) {
  int idx = blockIdx.x * 256 + threadIdx.x;   // 0 .. LAYERS*UNITS-1
  int l = idx >> 8;
  int n = idx & 255;
  bool oddl = (l & 1) != 0;
  // odd layers: D = roll(D_pre, +1)  =>  D[n] = D_pre[(n-1) mod 256]
  int src = oddl ? ((n + 255) & 255) : n;
  int m = src >> 1;          // MZI index
  int half = src & 1;        // 0 = top row, 1 = bottom row
  int mk = mask[l * MZIS + m];
  float th = mk ? theta[l * MZIS + m] : PI_F;
  float ph = mk ? phi[l * MZIS + m] : PI_F;
  float ur = __cosf(th), ui = __sinf(th);   // u = e^{i theta}
  float er = __cosf(ph), ei = __sinf(ph);   // e = e^{i phi}
  float dr, di, orr, oi;
  if (half == 0) {
    // d_top = 0.5 * e * (u - 1)
    float tr = ur - 1.0f, ti = ui;
    dr = 0.5f * (er * tr - ei * ti);
    di = 0.5f * (er * ti + ei * tr);
    // o_top = 0.5 * i * (u + 1) = 0.5 * (-ui, ur+1)
    orr = -0.5f * ui;
    oi  =  0.5f * (ur + 1.0f);
  } else {
    // d_bot = 0.5 * (1 - u)
    dr = 0.5f * (1.0f - ur);
    di = -0.5f * ui;
    // o_bot = 0.5 * i * e * (u + 1)
    float pr = er * (ur + 1.0f) - ei * ui;
    float pim = er * ui + ei * (ur + 1.0f);
    orr = -0.5f * pim;
    oi  =  0.5f * pr;
  }
  coef[idx] = make_float4(dr, di, orr, oi);
}

// ---------------------------------------------------------------------------
// Kernel 1: compose W = diag(e^{i gamma}) * A_1 * ... * A_L  (256x256 complex)
// One wave32 per row of W; each lane keeps 8 consecutive complex units in
// registers. Even layers pair (2k,2k+1): fully lane-local swap. Odd layers
// pair (2k+1,2k+2): only chunk boundaries cross lanes -> shfl; boundary lanes
// are safe because the rolled O is exactly 0 at units 0 and 255.
//
// Output is written directly in WMMA-B-ready "pair plane" layout:
//   Bre[kp*256 + n] = { W_re[2kp][n], W_re[2kp+1][n] }   (v2f, 8B)
//   Bim[kp*256 + n] = { W_im[2kp][n], W_im[2kp+1][n] }
// Row r of W is contraction index k=r; this wave writes component (r&1) of
// pair row kp = r>>1.
// ---------------------------------------------------------------------------
__global__ __launch_bounds__(256) void mesh_compose_kernel(
    const float* __restrict__ gamma, const float4* __restrict__ coef,
    float* __restrict__ BreF, float* __restrict__ BimF) {
  int warp = threadIdx.x >> 5;
  int lane = threadIdx.x & 31;
  int row = blockIdx.x * 8 + warp;   // 0..255
  int n0 = lane * 8;

  float g = gamma[row];
  float gr = __cosf(g), gi = __sinf(g);
  float2 y[8];
#pragma unroll
  for (int j = 0; j < 8; ++j) {
    int n = n0 + j;
    y[j].x = (n == row) ? gr : 0.0f;
    y[j].y = (n == row) ? gi : 0.0f;
  }

  for (int l = 0; l < LAYERS; ++l) {
    const float4* c = coef + l * UNITS + n0;
    float4 C[8];
#pragma unroll
    for (int j = 0; j < 8; ++j) C[j] = c[j];

    float2 p[8];
    if ((l & 1) == 0) {
#pragma unroll
      for (int j = 0; j < 8; ++j) p[j] = y[j ^ 1];
    } else {
      float2 prev7, next0;
      prev7.x = __shfl_up(y[7].x, 1, 32);
      prev7.y = __shfl_up(y[7].y, 1, 32);
      next0.x = __shfl_down(y[0].x, 1, 32);
      next0.y = __shfl_down(y[0].y, 1, 32);
      p[0] = prev7;                 // O=0 at global unit 0 -> lane0 garbage OK
      p[7] = next0;                 // O=0 at global unit 255 -> lane31 OK
#pragma unroll
      for (int j = 1; j < 7; ++j) p[j] = (j & 1) ? y[j + 1] : y[j - 1];
    }
#pragma unroll
    for (int j = 0; j < 8; ++j) {
      // y' = D*y + O*p   (complex)
      float tr = C[j].x * y[j].x - C[j].y * y[j].y
               + C[j].z * p[j].x - C[j].w * p[j].y;
      float ti = C[j].x * y[j].y + C[j].y * y[j].x
               + C[j].z * p[j].y + C[j].w * p[j].x;
      y[j].x = tr;
      y[j].y = ti;
    }
  }

  // scatter into pair-plane layout (component r&1 of pair row r>>1)
  int kp = row >> 1;
  int kb = row & 1;
  size_t base = ((size_t)kp * UNITS + n0) * 2 + kb;
#pragma unroll
  for (int j = 0; j < 8; ++j) {
    BreF[base + 2 * j] = y[j].x;
    BimF[base + 2 * j] = y[j].y;
  }
}

// ---------------------------------------------------------------------------
// Kernel 2: out[b,n] = sum_k x[b,k] * W[k,n]  with x real, W/out complex.
// Each wave computes a 16x32 strip (two 16x16 n-tiles, re+im accumulators =
// 4x v8f).  V_WMMA_F32_16X16X4_F32 operand layouts:
//   A (16x4, 2 VGPR): lane -> M = lane&15, K = 2*(lane>>4) + vgpr.
//   B (4x16, 2 VGPR): lane -> N = lane&15, K = 2*(lane>>4) + vgpr
//       -> exactly one v2f load from the pair plane at kp = k0/2 + (lane>>4).
//   D (16x16, 8 VGPR): VGPR r -> M = r + 8*(lane>>4), N = lane&15.
// Block = 8 waves covers all 256 columns of one 16-row m-tile (X rows stay
// hot in L1 across the block's waves).
// ---------------------------------------------------------------------------
__global__ __launch_bounds__(256) void mesh_gemm_kernel(
    const float* __restrict__ X, const v2f* __restrict__ Bre,
    const v2f* __restrict__ Bim, float2* __restrict__ out) {
  int warp = threadIdx.x >> 5;
  int lane = threadIdx.x & 31;
  int m0 = blockIdx.x * 16;   // 0..4080
  int n0 = warp * 32;         // two adjacent 16-wide n-tiles
  int li = lane & 15;         // M for A, N for B/D
  int kph = lane >> 4;        // which k-pair of the quad this half-wave owns

  v8f ar0 = {}, ai0 = {}, ar1 = {}, ai1 = {};
  const float* arow = X + (size_t)(m0 + li) * UNITS + 2 * kph;
  const v2f* brp = Bre + n0 + li;
  const v2f* bip = Bim + n0 + li;

  for (int k0 = 0; k0 < UNITS; k0 += 4) {
    size_t kp = (size_t)((k0 >> 1) + kph) * UNITS;
    v2f a   = *(const v2f*)(arow + k0);   // A[m][k], A[m][k+1]
    v2f br0 = brp[kp];                    // {Wre[k][n],   Wre[k+1][n]}
    v2f br1 = brp[kp + 16];               // n-tile 1
    v2f bi0 = bip[kp];
    v2f bi1 = bip[kp + 16];
    // 8 args: (neg_a, A, neg_b, B, c_mod, C, reuse_a, reuse_b)
    ar0 = __builtin_amdgcn_wmma_f32_16x16x4_f32(
        false, a, false, br0, (short)0, ar0, false, false);
    ai0 = __builtin_amdgcn_wmma_f32_16x16x4_f32(
        false, a, false, bi0, (short)0, ai0, false, false);
    ar1 = __builtin_amdgcn_wmma_f32_16x16x4_f32(
        false, a, false, br1, (short)0, ar1, false, false);
    ai1 = __builtin_amdgcn_wmma_f32_16x16x4_f32(
        false, a, false, bi1, (short)0, ai1, false, false);
  }

  int mbase = m0 + (kph << 3);
  int n = n0 + li;
  float2* orow = out + n;
#pragma unroll
  for (int r = 0; r < 8; ++r) {
    size_t off = (size_t)(mbase + r) * UNITS;
    float2 o0, o1;
    o0.x = ar0[r]; o0.y = ai0[r];
    o1.x = ar1[r]; o1.y = ai1[r];
    orow[off] = o0;
    orow[off + 16] = o1;
  }
}

// ---------------------------------------------------------------------------
// Launcher.  d_ws layout: [0, 256KB)      = Bre pair plane (128x256 v2f)
//                         [256KB, 512KB)  = Bim pair plane
//                         [512KB, 1.5MB)  = coef (256x256 float4)
// ---------------------------------------------------------------------------
extern "C" void kernel_launch(void* const* d_in, const int* in_sizes, int n_in,
                              void* d_out, int out_size, void* d_ws, size_t ws_size,
                              hipStream_t stream) {
  const float* x     = (const float*)d_in[0];   // [4096, 256]
  const float* theta = (const float*)d_in[1];   // [256, 128]
  const float* phi   = (const float*)d_in[2];   // [256, 128]
  const float* gamma = (const float*)d_in[3];   // [256]
  const int*   mask  = (const int*)d_in[4];     // [256, 128]

  const size_t plane = (size_t)(UNITS / 2) * UNITS * sizeof(v2f);  // 256 KB
  float* breF  = (float*)d_ws;
  float* bimF  = (float*)((char*)d_ws + plane);
  float4* coef = (float4*)((char*)d_ws + 2 * plane);
  float2* out  = (float2*)d_out;                // complex64 interleaved

  mesh_coef_kernel<<<(LAYERS * UNITS) / 256, 256, 0, stream>>>(theta, phi, mask, coef);
  mesh_compose_kernel<<<UNITS / 8, 256, 0, stream>>>(gamma, coef, breF, bimF);
  mesh_gemm_kernel<<<BATCH / 16, 256, 0, stream>>>(x, (const v2f*)breF,
                                                   (const v2f*)bimF, out);
}